// Model_65163243815318
// MI455X (gfx1250) — compile-verified
//
#include <hip/hip_runtime.h>

typedef __attribute__((ext_vector_type(2))) float v2f;
typedef __attribute__((ext_vector_type(8))) float v8f;

// ---- workspace layout (bytes) ----
// [0   .. 736)   double pw[23][4]   : pw[k] = M^(2^k), row-major 2x2
// [768 .. 1792)  float  T[256]      : T[r][j] = c_{r-1-j} (j<r), else 0
// [1792.. 1856)  float  G0[16]      : (H^T M^r)[0]
// [1856.. 1920)  float  G1[16]      : (H^T M^r)[1]
// [1920.. 1984)  float  W0[16]      : L*(M^{15-j} 1)[0]
// [1984.. 2048)  float  W1[16]      : L*(M^{15-j} 1)[1]
// [2048.. 2064)  float  P16[4]      : M^16
// [4096.. )      float2 v[nSuper]   : phase-B partial states, overwritten
//                                     in-place by phase C with x_start[]
#define WS_PW_OFF   0
#define WS_T_OFF    768
#define WS_G0_OFF   1792
#define WS_G1_OFF   1856
#define WS_W0_OFF   1920
#define WS_W1_OFF   1984
#define WS_P16_OFF  2048
#define WS_V_OFF    4096

// ---------------- kernel 1: build constants (1 thread) ----------------
__global__ void k_setup(const float* Ain, const float* Hin, const float* Lin,
                        void* ws) {
    if (threadIdx.x != 0 || blockIdx.x != 0) return;
    double* pw = (double*)((char*)ws + WS_PW_OFF);
    float* T   = (float*)((char*)ws + WS_T_OFF);
    float* G0  = (float*)((char*)ws + WS_G0_OFF);
    float* G1  = (float*)((char*)ws + WS_G1_OFF);
    float* W0  = (float*)((char*)ws + WS_W0_OFF);
    float* W1  = (float*)((char*)ws + WS_W1_OFF);
    float* P16 = (float*)((char*)ws + WS_P16_OFF);

    double H0 = (double)Hin[0], H1 = (double)Hin[1], Ld = (double)Lin[0];
    // M[i][j] = A[i][j] - L*H[j]
    double M00 = (double)Ain[0] - Ld * H0;
    double M01 = (double)Ain[1] - Ld * H1;
    double M10 = (double)Ain[2] - Ld * H0;
    double M11 = (double)Ain[3] - Ld * H1;

    pw[0] = M00; pw[1] = M01; pw[2] = M10; pw[3] = M11;
    for (int k = 1; k < 23; ++k) {
        double a = pw[(k-1)*4+0], b = pw[(k-1)*4+1];
        double c = pw[(k-1)*4+2], d = pw[(k-1)*4+3];
        pw[k*4+0] = a*a + b*c;  pw[k*4+1] = a*b + b*d;
        pw[k*4+2] = c*a + d*c;  pw[k*4+3] = c*b + d*d;
    }

    // R = M^i, iterate i = 0..15
    double R00 = 1, R01 = 0, R10 = 0, R11 = 1;
    double cc[16];
    for (int i = 0; i < 16; ++i) {
        G0[i] = (float)(H0*R00 + H1*R10);
        G1[i] = (float)(H0*R01 + H1*R11);
        double e0 = R00 + R01, e1 = R10 + R11;   // M^i * ones
        cc[i] = Ld * (H0*e0 + H1*e1);
        W0[15-i] = (float)(Ld * e0);
        W1[15-i] = (float)(Ld * e1);
        double n00 = M00*R00 + M01*R10, n01 = M00*R01 + M01*R11;
        double n10 = M10*R00 + M11*R10, n11 = M10*R01 + M11*R11;
        R00 = n00; R01 = n01; R10 = n10; R11 = n11;
    }
    for (int r = 0; r < 16; ++r)
        for (int j = 0; j < 16; ++j)
            T[r*16+j] = (j < r) ? (float)cc[r-1-j] : 0.0f;

    P16[0] = (float)pw[4*4+0]; P16[1] = (float)pw[4*4+1];
    P16[2] = (float)pw[4*4+2]; P16[3] = (float)pw[4*4+3];
}

// -------- kernel 2: per-super-tile local state (reference recurrence) ---
__global__ void k_local(const float* __restrict__ y,
                        const float* Ain, const float* Hin, const float* Lin,
                        float2* __restrict__ v, float* __restrict__ yhat,
                        int nSuper) {
    int t = blockIdx.x * blockDim.x + threadIdx.x;
    if (t == 0) yhat[0] = 0.0f;
    if (t >= nSuper) return;
    float A00 = Ain[0], A01 = Ain[1], A10 = Ain[2], A11 = Ain[3];
    float H0 = Hin[0], H1 = Hin[1], Lf = Lin[0];
    const float* yp = y + (size_t)t * 256;
    float x0 = 0.f, x1 = 0.f;
    for (int j = 0; j < 256; ++j) {
        float yv  = yp[j];
        float hx  = fmaf(H1, x1, H0 * x0);
        float inn = Lf * (yv - hx);
        float nx0 = fmaf(A01, x1, A00 * x0) + inn;
        float nx1 = fmaf(A11, x1, A10 * x0) + inn;
        x0 = nx0; x1 = nx1;
    }
    v[t] = make_float2(x0, x1);
}

// -------- kernel 3: scan over super-tile states (1 WG, f64) -----------
__global__ __launch_bounds__(1024)
void k_scan(float2* __restrict__ v, const void* ws, int nSuper) {
    __shared__ double sm0[1024];
    __shared__ double sm1[1024];
    const double* pw = (const double*)((const char*)ws + WS_PW_OFF);
    int tid = threadIdx.x;
    int per = nSuper >> 10;             // super-tiles per thread (pow2)
    int lp  = 31 - __clz(per);
    // P = M^256
    double P00 = pw[8*4+0], P01 = pw[8*4+1], P10 = pw[8*4+2], P11 = pw[8*4+3];
    int base = tid * per;
    double sx = 0.0, sy = 0.0;
    for (int i = 0; i < per; ++i) {
        float2 vv = v[base + i];
        double nx = P00*sx + P01*sy + (double)vv.x;
        double ny = P10*sx + P11*sy + (double)vv.y;
        sx = nx; sy = ny;
    }
    sm0[tid] = sx; sm1[tid] = sy;
    __syncthreads();
    for (int s = 0; (1 << s) < 1024; ++s) {
        // span of current accumulation = per*2^s items -> M^(2^(8+lp+s))
        const double* Q = pw + (size_t)(8 + lp + s) * 4;
        double px = 0.0, py = 0.0;
        int has = tid >= (1 << s);
        if (has) { px = sm0[tid - (1 << s)]; py = sm1[tid - (1 << s)]; }
        __syncthreads();
        if (has) {
            sx = sx + Q[0]*px + Q[1]*py;
            sy = sy + Q[2]*px + Q[3]*py;
            sm0[tid] = sx; sm1[tid] = sy;
        }
        __syncthreads();
    }
    double X0 = 0.0, X1 = 0.0;
    if (tid > 0) { X0 = sm0[tid-1]; X1 = sm1[tid-1]; }
    // replay: overwrite v[] in place with x_start for each super-tile
    sx = X0; sy = X1;
    for (int i = 0; i < per; ++i) {
        float2 vv = v[base + i];
        v[base + i] = make_float2((float)sx, (float)sy);
        double nx = P00*sx + P01*sy + (double)vv.x;
        double ny = P10*sx + P11*sy + (double)vv.y;
        sx = nx; sy = ny;
    }
}

// -------- kernel 4: WMMA Toeplitz multiply + in-wave 16-step scan ------
__global__ __launch_bounds__(256)
void k_emit(const float* __restrict__ y, float* __restrict__ yhat,
            const void* ws, const float2* __restrict__ xstart,
            int Ntot, int nSuper) {
    int lane = threadIdx.x & 31;
    int wid  = blockIdx.x * (blockDim.x >> 5) + (threadIdx.x >> 5);
    if (wid >= nSuper) return;          // wave-uniform branch

    const float* T   = (const float*)((const char*)ws + WS_T_OFF);
    const float* G0  = (const float*)((const char*)ws + WS_G0_OFF);
    const float* G1  = (const float*)((const char*)ws + WS_G1_OFF);
    const float* W0  = (const float*)((const char*)ws + WS_W0_OFF);
    const float* W1  = (const float*)((const char*)ws + WS_W1_OFF);
    const float* P16 = (const float*)((const char*)ws + WS_P16_OFF);

    int hi = lane >> 4;                 // half-wave selects K pair {2hi, 2hi+1}
    int m  = lane & 15;                 // A row / B column / tile index

    const float* yp = y + (size_t)wid * 256;

    // A operand: T[:,4k..4k+3] ; B operand: Y[4k..4k+3,:], Y[j][m]=yp[16m+j]
    v2f a[4], b[4];
#pragma unroll
    for (int k = 0; k < 4; ++k) {
        int j = 4*k + 2*hi;
        a[k].x = T[m*16 + j];
        a[k].y = T[m*16 + j + 1];
        b[k]   = *(const v2f*)(yp + 16*m + j);
    }

    v8f acc = {};
#pragma unroll
    for (int k = 0; k < 4; ++k) {
        acc = __builtin_amdgcn_wmma_f32_16x16x4_f32(
            false, a[k], false, b[k], (short)0, acc, false, false);
    }

    // u_m = W @ Y[:,m]  (each lane holds half the j's; combine across halves)
    float pu0 = 0.f, pu1 = 0.f;
#pragma unroll
    for (int k = 0; k < 4; ++k) {
        int j = 4*k + 2*hi;
        pu0 += W0[j]*b[k].x + W0[j+1]*b[k].y;
        pu1 += W1[j]*b[k].x + W1[j+1]*b[k].y;
    }
    float u0 = pu0 + __shfl_xor(pu0, 16);
    float u1 = pu1 + __shfl_xor(pu1, 16);

    // 16-step decimated scan: s_{m+1} = M^16 s_m + u_m
    float2 xs = xstart[wid];
    float sx = xs.x, sy = xs.y, sm0 = 0.f, sm1 = 0.f;
    float P00 = P16[0], P01 = P16[1], P10 = P16[2], P11 = P16[3];
#pragma unroll
    for (int mm = 0; mm < 16; ++mm) {
        float um0 = __shfl(u0, mm);
        float um1 = __shfl(u1, mm);
        if (mm == m) { sm0 = sx; sm1 = sy; }
        float nx = fmaf(P00, sx, fmaf(P01, sy, um0));
        float ny = fmaf(P10, sx, fmaf(P11, sy, um1));
        sx = nx; sy = ny;
    }

    // out_r = D[r][m] + G[r] . s_m ;  D lane layout: r = v + 8*hi, col = m
    size_t nbase = (size_t)wid * 256 + 16*(size_t)m + 8*(size_t)hi + 1;
#pragma unroll
    for (int vv = 0; vv < 8; ++vv) {
        int r = vv + 8*hi;
        float val = acc[vv] + fmaf(G0[r], sm0, G1[r] * sm1);
        size_t n = nbase + vv;
        if (n < (size_t)Ntot) yhat[n] = val;
    }
}

// ----------------------------- launcher -------------------------------
extern "C" void kernel_launch(void* const* d_in, const int* in_sizes, int n_in,
                              void* d_out, int out_size, void* d_ws, size_t ws_size,
                              hipStream_t stream) {
    const float* y = (const float*)d_in[0];
    const float* A = (const float*)d_in[1];
    const float* H = (const float*)d_in[2];
    const float* L = (const float*)d_in[3];
    float* yhat = (float*)d_out;
    int N = out_size;                    // 2^23
    int nSuper = N >> 8;                 // 256 steps per super-tile

    float2* v = (float2*)((char*)d_ws + WS_V_OFF);

    k_setup<<<1, 1, 0, stream>>>(A, H, L, d_ws);
    k_local<<<(nSuper + 255) / 256, 256, 0, stream>>>(y, A, H, L, v, yhat, nSuper);
    k_scan<<<1, 1024, 0, stream>>>(v, d_ws, nSuper);
    int blocks = (nSuper + 7) / 8;       // 8 waves (256 threads) per block
    k_emit<<<blocks, 256, 0, stream>>>(y, yhat, d_ws, v, N, nSuper);
}